// NMSLoss4_87136296501789
// MI455X (gfx1250) — compile-verified
//
#include <hip/hip_runtime.h>
#include <math.h>
#include <stdint.h>

#define B_IMG 8
#define N_PROP 2048
#define G_GT 64
#define NMS_THR_F 0.5f
#define MIN_HEIGHT_F 50.0f
#define EPS_F 1e-6f
#define TPB 256
#define NWAVE (TPB / 32)
#define PER (N_PROP / TPB) // 8 elements per thread

// ---- CDNA5 async global->LDS staging (ASYNCcnt path), inline asm so it is
// portable across ROCm7.2/clang-22 and the amdgpu-toolchain builtin arities.
__device__ __forceinline__ uint32_t lds_off_u32(const void* p) {
    // AS3 -> generic cast keeps the LDS byte offset in the low 32 bits.
    return (uint32_t)(uint64_t)p;
}

__device__ __forceinline__ void async_gl2lds_b32(uint32_t lds_byte_off,
                                                 uint32_t g_byte_off,
                                                 const void* g_base) {
    asm volatile("global_load_async_to_lds_b32 %0, %1, %2"
                 :
                 : "v"(lds_byte_off), "v"(g_byte_off), "s"(g_base)
                 : "memory");
}

__device__ __forceinline__ void async_gl2lds_b128(uint32_t lds_byte_off,
                                                  uint32_t g_byte_off,
                                                  const void* g_base) {
    asm volatile("global_load_async_to_lds_b128 %0, %1, %2"
                 :
                 : "v"(lds_byte_off), "v"(g_byte_off), "s"(g_base)
                 : "memory");
}

__device__ __forceinline__ void wait_asynccnt0() {
    asm volatile("s_wait_asynccnt 0" ::: "memory");
}

// box = (x1, y1, x2, y2) in (x, y, z, w)
__device__ __forceinline__ float iou_f4(float4 a, float4 b) {
    float areaA = (a.z - a.x) * (a.w - a.y);
    float areaB = (b.z - b.x) * (b.w - b.y);
    float ltx = fmaxf(a.x, b.x), lty = fmaxf(a.y, b.y);
    float rbx = fminf(a.z, b.z), rby = fminf(a.w, b.w);
    float w = fmaxf(rbx - ltx, 0.0f);
    float h = fmaxf(rby - lty, 0.0f);
    float inter = w * h;
    return inter / (areaA + areaB - inter + 1e-10f);
}

__global__ void __launch_bounds__(TPB) nms_loss_per_image(
    const int* __restrict__ agti_all,   // (B, N) int32
    const float* __restrict__ gtb_all,  // (B, G, 4)
    const float* __restrict__ prop_all, // (B, N, 5)
    float* __restrict__ ws)             // (B, 2): [push, pull] per image
{
    __shared__ float4 sBox[N_PROP];     // 32 KB, one ds_load_b128 per IoU operand
    __shared__ float sS[N_PROP];        // 8 KB
    __shared__ int sAgti[N_PROP];       // 8 KB
    __shared__ int sActive[N_PROP];     // 8 KB
    __shared__ float4 sGt[G_GT];        // 1 KB
    __shared__ unsigned sSeen[2];
    __shared__ float sWS[NWAVE];
    __shared__ int sWI[NWAVE];
    __shared__ float sPushV[G_GT];
    __shared__ int sPushC[G_GT];
    __shared__ int sActiveCount;
    __shared__ int sPosCount;

    const int b = blockIdx.x;
    const int tid = threadIdx.x;
    const int lane = tid & 31;
    const int wid = tid >> 5;

    const float* prop = prop_all + (size_t)b * N_PROP * 5;
    const int* agti = agti_all + (size_t)b * N_PROP;
    const float* gtb = gtb_all + (size_t)b * G_GT * 4;

    // ---- Stage proposals into LDS via async global->LDS (ASYNCcnt) ----
    // box: one b128 (prop stride is 20B; DWORD-aligned b128 is legal in DWORD mode)
    // score + agti: one b32 each.
    for (int k = 0; k < PER; ++k) {
        int j = tid + k * TPB;
        uint32_t gb = (uint32_t)(j * 5 * 4);
        async_gl2lds_b128(lds_off_u32(&sBox[j]), gb, prop);
        async_gl2lds_b32(lds_off_u32(&sS[j]), gb + 16u, prop);
        async_gl2lds_b32(lds_off_u32(&sAgti[j]), (uint32_t)(j * 4), agti);
    }
    if (tid < G_GT) {
        sGt[tid] = ((const float4*)gtb)[tid];
    }
    if (tid == 0) {
        sSeen[0] = 0u;
        sSeen[1] = 0u;
        sPosCount = 0;
    }
    wait_asynccnt0();
    __syncthreads();

    // ---- Init active set from agti >= 0, count positives ----
    int cnt = 0;
    for (int k = 0; k < PER; ++k) {
        int j = tid + k * TPB;
        int act = (sAgti[j] >= 0) ? 1 : 0;
        sActive[j] = act;
        cnt += act;
    }
    // wave-reduce then one atomic per wave
    for (int off = 16; off > 0; off >>= 1) cnt += __shfl_down(cnt, off, 32);
    if (lane == 0 && cnt) atomicAdd(&sPosCount, cnt);
    __syncthreads();
    if (tid == 0) sActiveCount = sPosCount;

    float tp = 0.0f; // live in thread 0 only
    int pc = 0;      // live in thread 0 only

    // ---- Sequential NMS scan: 2 barriers per step ----
    for (int step = 0; step < N_PROP; ++step) {
        __syncthreads(); // (A) publishes removals + sActiveCount of prior step
        int ac = sActiveCount;
        if (ac == 0) break;

        // per-thread argmax over its strided slice (strict > => first index wins)
        float bs = -INFINITY;
        int bi = 0x7FFFFFFF;
        for (int k = 0; k < PER; ++k) {
            int j = tid + k * TPB;
            if (sActive[j]) {
                float s = sS[j];
                if (s > bs || (s == bs && j < bi)) { bs = s; bi = j; }
            }
        }
        // wave32 shuffle reduction
        for (int off = 16; off > 0; off >>= 1) {
            float os = __shfl_down(bs, off, 32);
            int oi = __shfl_down(bi, off, 32);
            if (os > bs || (os == bs && oi < bi)) { bs = os; bi = oi; }
        }
        if (lane == 0) { sWS[wid] = bs; sWI[wid] = bi; }
        __syncthreads(); // (B) partials visible

        // every thread merges the 8 partials (broadcast LDS reads, no 3rd barrier)
        float fb = sWS[0];
        int fi = sWI[0];
        for (int w = 1; w < NWAVE; ++w) {
            float os = sWS[w];
            int oi = sWI[w];
            if (os > fb || (os == fb && oi < fi)) { fb = os; fi = oi; }
        }
        const int i = fi;

        // thread 0: pull-loss bookkeeping (disjoint LDS from suppression pass)
        if (tid == 0) {
            int g = sAgti[i];
            if (g < 0) g = 0;
            unsigned seen_g = (sSeen[g >> 5] >> (g & 31)) & 1u;
            if (seen_g) {
                pc += 1;      // do_pull does NOT require has_rem
                if (ac > 1) { // tp does require has_rem
                    float msi = fmaxf(iou_f4(sGt[g], sBox[i]), EPS_F);
                    float pull = -logf(fminf(1.0f - NMS_THR_F + msi, 1.0f)) * sS[i];
                    tp += pull;
                }
            }
            sSeen[g >> 5] |= (1u << (g & 31));
        }

        // suppression: remove i and all actives with IoU(i, j) > 0.5
        float4 boxI = sBox[i];
        int removed = 0;
        for (int k = 0; k < PER; ++k) {
            int j = tid + k * TPB;
            if (sActive[j]) {
                if (j == i) {
                    sActive[j] = 0;
                    removed++;
                } else if (iou_f4(boxI, sBox[j]) > NMS_THR_F) {
                    sActive[j] = 0;
                    removed++;
                }
            }
        }
        for (int off = 16; off > 0; off >>= 1) removed += __shfl_down(removed, off, 32);
        if (lane == 0 && removed) atomicSub(&sActiveCount, removed);
    }
    __syncthreads();

    // ---- Push loss: best proposal per GT ----
    if (tid < G_GT) {
        int g = tid;
        float bsc = -INFINITY;
        int bj = -1;
        for (int j = 0; j < N_PROP; ++j) {
            if (sAgti[j] == g) { // implies pos (agti >= 0)
                float s = sS[j];
                if (s > bsc) { bsc = s; bj = j; } // strict > => first-index argmax
            }
        }
        float4 gt = sGt[g];
        float h = gt.w - gt.y;
        unsigned seen_g = (sSeen[g >> 5] >> (g & 31)) & 1u;
        int pm = (bj >= 0) && (h >= MIN_HEIGHT_F) && (!seen_g);
        float pv = 0.0f;
        if (pm) {
            pv = 1.0f - iou_f4(gt, sBox[bj]);
        }
        sPushV[g] = pv;
        sPushC[g] = pm;
    }
    __syncthreads();

    if (tid == 0) {
        float tot = 0.0f;
        int cpush = 0;
        for (int g = 0; g < G_GT; ++g) {
            tot += sPushV[g];
            cpush += sPushC[g];
        }
        int valid = (sPosCount > 1);
        float pull_loss = valid ? (tp / ((float)pc + EPS_F)) : 0.0f;
        float push_loss = valid ? (tot / ((float)cpush + EPS_F)) : 0.0f;
        ws[b * 2 + 0] = push_loss;
        ws[b * 2 + 1] = pull_loss;
    }
}

__global__ void nms_loss_reduce(const float* __restrict__ ws, float* __restrict__ out) {
    if (threadIdx.x == 0 && blockIdx.x == 0) {
        float sp = 0.0f, sl = 0.0f;
        for (int b = 0; b < B_IMG; ++b) {
            sp += ws[b * 2 + 0];
            sl += ws[b * 2 + 1];
        }
        out[0] = (sp / (float)B_IMG) * 1.0f; // PUSH_W
        out[1] = (sl / (float)B_IMG) * 1.0f; // PULL_W
    }
}

extern "C" void kernel_launch(void* const* d_in, const int* in_sizes, int n_in,
                              void* d_out, int out_size, void* d_ws, size_t ws_size,
                              hipStream_t stream) {
    (void)in_sizes; (void)n_in; (void)out_size; (void)ws_size;
    // setup_inputs() order: gt_inds (unused), anchor_gt_inds, gt_bboxes, proposal_list
    const int* agti = (const int*)d_in[1];
    const float* gtb = (const float*)d_in[2];
    const float* prop = (const float*)d_in[3];
    float* ws = (float*)d_ws;
    float* out = (float*)d_out;

    nms_loss_per_image<<<B_IMG, TPB, 0, stream>>>(agti, gtb, prop, ws);
    nms_loss_reduce<<<1, 32, 0, stream>>>(ws, out);
}